// XceptionResNet_24739011625841
// MI455X (gfx1250) — compile-verified
//
#include <hip/hip_runtime.h>

// Problem constants (fixed by the reference)
#define CC 64          // channels
#define KK 27          // 3x3x3 kernel taps
#define LDS_STRIDE 66  // padded stride for transposed W in LDS (bank-conflict free)
#define IDX_STRIDE 28  // padded stride for per-row neighbor indices in LDS

typedef __attribute__((ext_vector_type(2))) float v2f;
typedef __attribute__((ext_vector_type(8))) float v8f;

// ---------------------------------------------------------------------------
// Kernel 1: bx = feat @ W + b1   via V_WMMA_F32_16X16X4_F32
// Block = 256 threads = 8 waves. Each wave computes a 16x64 output tile,
// stepping K by 4 -> 16 K-steps x 4 N-tiles = 64 WMMAs per wave.
// (Unchanged from previous round: disasm shows clean WMMA/VMEM/DS overlap.)
// ---------------------------------------------------------------------------
__global__ __launch_bounds__(256)
void gemm_bias_wmma(const float* __restrict__ feat,
                    const float* __restrict__ W,
                    const float* __restrict__ b1,
                    float* __restrict__ bx,
                    int Npts)
{
    // W transposed into LDS: sWt[n*66 + k] = W[k*64 + n]
    // -> B fragment (v0,v1 = B[k][n], B[k+1][n]) is one aligned ds_load_b64,
    //    and the 66-float row stride spreads the 16 lanes over distinct banks.
    __shared__ float sWt[CC * LDS_STRIDE];
    for (int i = threadIdx.x; i < CC * CC; i += 256) {
        int k = i >> 6;
        int n = i & 63;
        sWt[n * LDS_STRIDE + k] = W[i];
    }
    __syncthreads();

    const int lane = threadIdx.x & 31;
    const int wave = threadIdx.x >> 5;
    const int l15  = lane & 15;
    const int hs   = lane >> 4;                 // half-select (K pair 0/1 vs 2/3)
    const int rowBase = (blockIdx.x * 8 + wave) * 16;

    // Clamp A-row reads so a ragged last block stays in bounds WITHOUT
    // diverging EXEC before the WMMAs (EXEC must be all-1s for WMMA).
    int aRowIdx = rowBase + l15;
    if (aRowIdx > Npts - 1) aRowIdx = Npts - 1;
    const float* aRow = feat + (size_t)aRowIdx * CC;

    v8f acc[4] = {v8f{}, v8f{}, v8f{}, v8f{}};

#pragma unroll
    for (int k0 = 0; k0 < CC; k0 += 4) {
        // A fragment: 16x4 f32. lane l<16: (M=l, K=k0..k0+1); lanes 16-31: K=k0+2..k0+3
        v2f a = *(const v2f*)(aRow + k0 + hs * 2);
#pragma unroll
        for (int nt = 0; nt < 4; ++nt) {
            // B fragment: 4x16 f32. lane l<16: (K=k0..k0+1, N=nt*16+l); lanes 16-31: K=k0+2..k0+3
            const float* bp = &sWt[(nt * 16 + l15) * LDS_STRIDE + k0 + hs * 2];
            v2f b = *(const v2f*)bp;
            acc[nt] = __builtin_amdgcn_wmma_f32_16x16x4_f32(
                false, a, false, b, (short)0, acc[nt], false, false);
        }
    }

    // C/D layout: VGPR r holds rows M=r (lanes 0-15) and M=8+r (lanes 16-31).
#pragma unroll
    for (int nt = 0; nt < 4; ++nt) {
        const int col = nt * 16 + l15;
        const float bias = b1[col];
#pragma unroll
        for (int r = 0; r < 8; ++r) {
            const int row = rowBase + hs * 8 + r;
            if (row < Npts) {
                bx[(size_t)row * CC + col] = acc[nt][r] + bias;
            }
        }
    }
}

// ---------------------------------------------------------------------------
// Kernel 2: out = feat + b2 + sum_k mask * bx[nb[k]] * dw[k]
// Block = 256 threads = 16 rows x 16 four-channel groups.
//  - The block's 16x27 neighbor-index slab (contiguous in memory) is staged
//    into LDS with one coalesced pass, then read per-tap as an LDS broadcast
//    (padded stride 28) instead of 16 redundant global loads per tap.
//  - dw (27x64 f32, 6.75 KB) is preloaded into LDS -> per-tap weight read is
//    a ds_load_b128, so the only VMEM op in the hot loop is the bx gather.
//  - 16 consecutive lanes share a point, so the `idx >= 0` branch is uniform
//    per half-wave; at ~12.5% grid occupancy ~87% of taps are skipped.
//  - bx (64 MB) fits in the 192 MB L2 -> surviving gathers are L2-bound.
// ---------------------------------------------------------------------------
__global__ __launch_bounds__(256)
void dw_gather_residual(const float* __restrict__ feat,
                        const int*   __restrict__ nb,
                        const float* __restrict__ bx,
                        const float* __restrict__ dw,
                        const float* __restrict__ b2,
                        float*       __restrict__ out,
                        int Npts)
{
    __shared__ int   sIdx[16 * IDX_STRIDE];   // 16 rows x 27 taps, padded
    __shared__ float sDw[KK * CC];            // tap weights

    const int rowTile = blockIdx.x * 16;

    // Coalesced stage of the contiguous index slab nb[rowTile*27 .. +16*27)
    {
        int validRows = Npts - rowTile;
        if (validRows > 16) validRows = 16;
        const int total = validRows * KK;
        const int* slab = nb + (size_t)rowTile * KK;
        for (int i = threadIdx.x; i < total; i += 256) {
            // i = r*27 + k
            const int r = i / KK;
            const int k = i - r * KK;
            sIdx[r * IDX_STRIDE + k] = slab[i];
        }
    }
    // Preload dw into LDS (1728 floats)
    for (int i = threadIdx.x; i < KK * CC; i += 256) {
        sDw[i] = dw[i];
    }
    __syncthreads();

    const int r   = threadIdx.x >> 4;         // row within tile (0..15)
    const int c4  = (threadIdx.x & 15) << 2;  // channel group * 4
    const int row = rowTile + r;
    if (row >= Npts) return;

    const int* myIdx = &sIdx[r * IDX_STRIDE];

    float ax = 0.f, ay = 0.f, az = 0.f, aw = 0.f;
#pragma unroll
    for (int k = 0; k < KK; ++k) {
        const int idx = myIdx[k];              // LDS broadcast within row group
        if (idx >= 0) {
            const float4 g = *(const float4*)(bx + (size_t)idx * CC + c4);
            const float4 w = *(const float4*)(&sDw[k * CC + c4]);
            ax += g.x * w.x;
            ay += g.y * w.y;
            az += g.z * w.z;
            aw += g.w * w.w;
        }
    }

    const float4 f  = *(const float4*)(feat + (size_t)row * CC + c4);
    const float4 bb = *(const float4*)(b2 + c4);
    float4 o;
    o.x = f.x + ax + bb.x;
    o.y = f.y + ay + bb.y;
    o.z = f.z + az + bb.z;
    o.w = f.w + aw + bb.w;
    *(float4*)(out + (size_t)row * CC + c4) = o;
}

// ---------------------------------------------------------------------------
// Launch. Inputs (setup_inputs order): feat, neighbor_idx, W, b1, b2, dw.
// d_ws holds bx ([N,64] f32 = 64 MB).
// ---------------------------------------------------------------------------
extern "C" void kernel_launch(void* const* d_in, const int* in_sizes, int n_in,
                              void* d_out, int out_size, void* d_ws, size_t ws_size,
                              hipStream_t stream)
{
    const float* feat = (const float*)d_in[0];
    const int*   nb   = (const int*)  d_in[1];
    const float* W    = (const float*)d_in[2];
    const float* b1   = (const float*)d_in[3];
    const float* b2   = (const float*)d_in[4];
    const float* dw   = (const float*)d_in[5];
    float* out = (float*)d_out;
    float* bx  = (float*)d_ws;   // [N, 64] f32 scratch

    const int Npts = in_sizes[0] / CC;

    // GEMM: 128 rows per block (8 waves x 16 rows)
    const int gemmBlocks = (Npts + 127) / 128;
    gemm_bias_wmma<<<gemmBlocks, 256, 0, stream>>>(feat, W, b1, bx, Npts);

    // Depthwise gather + residual: 16 rows per block
    const int dwBlocks = (Npts + 15) / 16;
    dw_gather_residual<<<dwBlocks, 256, 0, stream>>>(feat, nb, bx, dw, b2, out, Npts);
}